// GraphConvNet_86320252715499
// MI455X (gfx1250) — compile-verified
//
#include <hip/hip_runtime.h>
#include <hip/hip_bf16.h>

// ---------------------------------------------------------------------------
// GraphConvNet on MI455X (gfx1250):
//  - fp32 WMMA (V_WMMA_F32_16X16X4_F32) for the 128x128 MLP GEMMs
//  - async global->LDS copies (GLOBAL_LOAD_ASYNC_TO_LDS_B32) stage W with an
//    on-the-fly pair-interleave repack (scattered per-lane LDS destinations),
//    so B fragments are single ds_load_b64s
//  - TDM (tensor_load_to_lds) DMAs the bias row into LDS
//  - fused bias+ReLU(+inv_sqrt_s) epilogue, wave-per-edge atomic scatter,
//    wave-per-node LayerNorm with wave32 shuffles
// ---------------------------------------------------------------------------

typedef __attribute__((ext_vector_type(2))) float v2f;
typedef __attribute__((ext_vector_type(4))) float v4f;
typedef __attribute__((ext_vector_type(8))) float v8f;
typedef __attribute__((ext_vector_type(4))) unsigned int u32x4;
typedef __attribute__((ext_vector_type(8))) unsigned int u32x8;

#define LATENT 128
#define IN_FEAT 7
#define STEPS 3

// ---------------------------------------------------------------------------
// Utility: zero-fill a float buffer
// ---------------------------------------------------------------------------
__global__ void gcn_fill_zero(float* __restrict__ p, int n) {
    int i = blockIdx.x * blockDim.x + threadIdx.x;
    if (i < n) p[i] = 0.0f;
}

// ---------------------------------------------------------------------------
// Degree accumulation: deg[s] += 1, deg[r] += 1
// ---------------------------------------------------------------------------
__global__ void gcn_deg_accum(const int* __restrict__ snd, const int* __restrict__ rcv,
                              float* __restrict__ degS, float* __restrict__ degR, int nE) {
    int e = blockIdx.x * blockDim.x + threadIdx.x;
    if (e < nE) {
        atomicAdd(&degS[snd[e]], 1.0f);
        atomicAdd(&degR[rcv[e]], 1.0f);
    }
}

// In-place: deg -> rsqrt(max(deg, 1))
__global__ void gcn_inv_sqrt(float* __restrict__ deg, int n) {
    int i = blockIdx.x * blockDim.x + threadIdx.x;
    if (i < n) deg[i] = rsqrtf(fmaxf(deg[i], 1.0f));
}

// ---------------------------------------------------------------------------
// Embed: h[i,j] = b[j] + sum_k nodes[i,k] * We[k,j]   (7 -> 128)
// ---------------------------------------------------------------------------
__global__ void gcn_embed(const float* __restrict__ nodes, const float* __restrict__ We,
                          const float* __restrict__ be, float* __restrict__ h, int n) {
    int idx = blockIdx.x * blockDim.x + threadIdx.x;
    if (idx >= n * LATENT) return;
    int i = idx >> 7;
    int j = idx & (LATENT - 1);
    float acc = be[j];
#pragma unroll
    for (int k = 0; k < IN_FEAT; ++k)
        acc += nodes[i * IN_FEAT + k] * We[k * LATENT + j];
    h[idx] = acc;
}

// ---------------------------------------------------------------------------
// WMMA GEMM:  Y = relu(X @ W + b) [* rowScale]
//   X: [n,128], W: [128,128] row-major (K x N), b: [128], Y: [n,128]
//
// 256 threads = 8 waves; wave w computes rows [blk*128 + w*16, +16), all 128
// cols as 8 WMMA 16x16 tiles, K-loop of 32 chunks of 4 (fp32 WMMA).
//
// W is staged into LDS pair-packed: lds[((k>>1)*128 + n)*2 + (k&1)], via
// GLOBAL_LOAD_ASYNC_TO_LDS_B32 (per-lane scattered LDS destinations do the
// repack in flight; ASYNCcnt tracked). Bias row staged by the TDM with one
// tensor_load_to_lds (TENSORcnt tracked).
//
// Fragment layouts (CDNA5 ISA 7.12.2, wave32):
//   A(16x4 f32):  lane = m + 16*kpair, vgpr = k%2 -> lane holds A[m][2*kpair+{0,1}]
//   B(4x16 f32):  lane = n + 16*kpair, vgpr = k%2 -> lane holds B[2*kpair+{0,1}][n]
//   C/D(16x16):   vgpr r, lane l -> element (m = r + 8*(l/16), n = l%16)
// ---------------------------------------------------------------------------
__global__ __launch_bounds__(256)
void gcn_mlp_gemm_relu(const float* __restrict__ X, const float* __restrict__ W,
                       const float* __restrict__ bias, const float* __restrict__ rowScale,
                       float* __restrict__ Y, int n) {
    __shared__ float lw[LATENT * LATENT];  // 64 KB pair-packed W (of 320 KB/WGP)
    __shared__ float lbias[LATENT];        // 512 B bias, DMAed by TDM

    const int t = threadIdx.x;

    // --- TDM: DMA the bias row (1 x 128 fp32) into LDS (wave 0 only; TDM
    // ignores EXEC and issues once per wave). D# per ISA 08 §8.3/8.4.
    if (t < 32) {
        unsigned long long gaddr = (unsigned long long)(const void*)bias;
        unsigned int ldsaddr = (unsigned int)(unsigned long long)(void*)&lbias[0];

        u32x4 g0;
        g0[0] = 1u;                                   // count=1, user mode
        g0[1] = ldsaddr;                              // lds_addr (bytes)
        g0[2] = (unsigned int)(gaddr & 0xFFFFFFFFu);  // global_addr[31:0]
        g0[3] = (unsigned int)((gaddr >> 32) & 0x01FFFFFFu) | 0x80000000u; // [56:32] | type=2

        u32x8 g1;
        g1[0] = 2u << 16;          // workgroup_mask=0, data_size=2 (4B)
        g1[1] = 128u << 16;        // tensor_dim0 = 128
        g1[2] = 1u << 16;          // tensor_dim1 = 1
        g1[3] = 128u << 16;        // tile_dim0  = 128
        g1[4] = 1u;                // tile_dim1  = 1; tile_dim2 = 0
        g1[5] = 128u;              // tensor_dim0_stride = 128
        g1[6] = 0u;
        g1[7] = 0u;

        asm volatile("tensor_load_to_lds %0, %1" :: "s"(g0), "s"(g1) : "memory");
    }

    // --- Async global->LDS staging of W with pair-interleave repack.
    // Thread t copies elements idx = t + 256*i  (k = idx>>7, ncol = idx&127).
    // Packed LDS byte address:  base + 8*ncol + 4*k0 + 1024*i   (k0 = t>>7)
    // Global byte address:      W + 4*t + 1024*i
    {
        unsigned long long g = (unsigned long long)(const void*)W + 4ull * (unsigned)t;
        unsigned int ldsa = (unsigned int)(unsigned long long)(void*)&lw[0]
                          + 8u * (unsigned)(t & 127) + 4u * (unsigned)(t >> 7);
#pragma unroll 4
        for (int i = 0; i < (LATENT * LATENT) / 256; ++i) {
            asm volatile("global_load_async_to_lds_b32 %0, %1, off"
                         :: "v"(ldsa), "v"(g) : "memory");
            ldsa += 1024u;
            g += 1024ull;
        }
    }
    asm volatile("s_wait_asynccnt 0x0" ::: "memory");
    if (t < 32) __builtin_amdgcn_s_wait_tensorcnt(0);
    __syncthreads();

    const int wave  = t >> 5;
    const int lane  = t & 31;
    const int lhalf = lane >> 4;  // which k-pair of the 4-chunk this lane holds
    const int l16   = lane & 15;

    const int rowBase = blockIdx.x * 128 + wave * 16;

    // A-row for this lane (clamped: EXEC must stay all-1s for WMMA; mask at store)
    int arow = rowBase + l16;
    if (arow >= n) arow = n - 1;
    const float* aptr = X + (size_t)arow * LATENT + (lhalf << 1);

    v8f acc[8] = {};

    for (int kc = 0; kc < 32; ++kc) {
        v2f a = *(const v2f*)(aptr + kc * 4);
        const int kbase = kc * 4 + (lhalf << 1);                     // even
        const float* bbase = lw + (((kbase >> 1) << 7) + l16) * 2;   // +ntile*32
#pragma unroll
        for (int ntile = 0; ntile < 8; ++ntile) {
            v2f b = *(const v2f*)(bbase + ntile * 32);   // ds_load_b64
            acc[ntile] = __builtin_amdgcn_wmma_f32_16x16x4_f32(
                /*neg_a=*/false, a, /*neg_b=*/false, b,
                /*c_mod=*/(short)0, acc[ntile],
                /*reuse_a=*/false, /*reuse_b=*/false);
        }
    }

    // Epilogue: bias + ReLU (+ optional per-row scale), masked store
#pragma unroll
    for (int ntile = 0; ntile < 8; ++ntile) {
        const int col = ntile * 16 + l16;
        const float bcol = lbias[col];
#pragma unroll
        for (int r = 0; r < 8; ++r) {
            const int row = rowBase + r + (lhalf << 3);
            if (row < n) {
                float v = acc[ntile][r] + bcol;
                v = v > 0.0f ? v : 0.0f;
                if (rowScale) v *= rowScale[row];
                Y[(size_t)row * LATENT + col] = v;
            }
        }
    }
}

// ---------------------------------------------------------------------------
// Edge scatter: agg[rcv[e], :] += x[snd[e], :]   (one wave per edge,
// each lane moves a float4 via 4 fp32 atomics; agg is L2-resident)
// ---------------------------------------------------------------------------
__global__ __launch_bounds__(256)
void gcn_scatter(const float* __restrict__ x, const int* __restrict__ snd,
                 const int* __restrict__ rcv, float* __restrict__ agg, int nE) {
    int e = blockIdx.x * 8 + (threadIdx.x >> 5);
    if (e >= nE) return;
    int lane = threadIdx.x & 31;
    int s = snd[e];
    int r = rcv[e];
    v4f xv = *(const v4f*)(x + (size_t)s * LATENT + lane * 4);
    float* dst = agg + (size_t)r * LATENT + lane * 4;
    atomicAdd(dst + 0, xv.x);
    atomicAdd(dst + 1, xv.y);
    atomicAdd(dst + 2, xv.z);
    atomicAdd(dst + 3, xv.w);
}

// ---------------------------------------------------------------------------
// Skip + LayerNorm: h = LN(h + agg * inv_sqrt_r) * scale + bias
// One wave per node, wave32 shuffle reduction over 128 features (4/lane).
// ---------------------------------------------------------------------------
__global__ __launch_bounds__(256)
void gcn_update_ln(float* __restrict__ h, const float* __restrict__ agg,
                   const float* __restrict__ invR, const float* __restrict__ lnS,
                   const float* __restrict__ lnB, int n) {
    int node = blockIdx.x * 8 + (threadIdx.x >> 5);
    if (node >= n) return;
    int lane = threadIdx.x & 31;
    size_t base = (size_t)node * LATENT + lane * 4;

    v4f hv = *(const v4f*)(h + base);
    v4f av = *(const v4f*)(agg + base);
    float is = invR[node];
    float v0 = hv.x + av.x * is;
    float v1 = hv.y + av.y * is;
    float v2 = hv.z + av.z * is;
    float v3 = hv.w + av.w * is;

    float s  = v0 + v1 + v2 + v3;
    float sq = v0 * v0 + v1 * v1 + v2 * v2 + v3 * v3;
#pragma unroll
    for (int off = 16; off >= 1; off >>= 1) {
        s  += __shfl_xor(s, off, 32);
        sq += __shfl_xor(sq, off, 32);
    }
    float mean = s * (1.0f / (float)LATENT);
    float var  = sq * (1.0f / (float)LATENT) - mean * mean;
    float rstd = rsqrtf(var + 1e-6f);

    int c = lane * 4;
    h[base + 0] = (v0 - mean) * rstd * lnS[c + 0] + lnB[c + 0];
    h[base + 1] = (v1 - mean) * rstd * lnS[c + 1] + lnB[c + 1];
    h[base + 2] = (v2 - mean) * rstd * lnS[c + 2] + lnB[c + 2];
    h[base + 3] = (v3 - mean) * rstd * lnS[c + 3] + lnB[c + 3];
}

// ---------------------------------------------------------------------------
// Decode: out[i,c] = b[c] + sum_j h[i,j] * Wd[j,c]   (128 -> 7)
// One wave per node; per-lane partial dots + wave32 shuffle reduce.
// ---------------------------------------------------------------------------
__global__ __launch_bounds__(256)
void gcn_decode(const float* __restrict__ h, const float* __restrict__ Wd,
                const float* __restrict__ bd, float* __restrict__ out, int n) {
    int node = blockIdx.x * 8 + (threadIdx.x >> 5);
    if (node >= n) return;
    int lane = threadIdx.x & 31;

    v4f hv = *(const v4f*)(h + (size_t)node * LATENT + lane * 4);
    float acc[IN_FEAT] = {0.f, 0.f, 0.f, 0.f, 0.f, 0.f, 0.f};
    int jb = lane * 4;
#pragma unroll
    for (int jj = 0; jj < 4; ++jj) {
        float hvj = hv[jj];
#pragma unroll
        for (int c = 0; c < IN_FEAT; ++c)
            acc[c] += hvj * Wd[(jb + jj) * IN_FEAT + c];
    }
#pragma unroll
    for (int c = 0; c < IN_FEAT; ++c) {
#pragma unroll
        for (int off = 16; off >= 1; off >>= 1)
            acc[c] += __shfl_xor(acc[c], off, 32);
    }
    if (lane == 0) {
#pragma unroll
        for (int c = 0; c < IN_FEAT; ++c)
            out[(size_t)node * IN_FEAT + c] = acc[c] + bd[c];
    }
}

// ---------------------------------------------------------------------------
// Launch
// ---------------------------------------------------------------------------
extern "C" void kernel_launch(void* const* d_in, const int* in_sizes, int n_in,
                              void* d_out, int out_size, void* d_ws, size_t ws_size,
                              hipStream_t stream) {
    (void)n_in; (void)out_size; (void)ws_size;

    const float* nodes   = (const float*)d_in[0];
    const int*   senders = (const int*)d_in[1];
    const int*   recvers = (const int*)d_in[2];
    const float* W_embed = (const float*)d_in[3];
    const float* b_embed = (const float*)d_in[4];
    const float* mlp_W   = (const float*)d_in[5];
    const float* mlp_b   = (const float*)d_in[6];
    const float* ln_scale= (const float*)d_in[7];
    const float* ln_bias = (const float*)d_in[8];
    const float* W_dec   = (const float*)d_in[9];
    const float* b_dec   = (const float*)d_in[10];
    float* out = (float*)d_out;

    const int n  = in_sizes[0] / IN_FEAT;  // 50000
    const int nE = in_sizes[1];            // 600000
    const size_t NL = (size_t)n * LATENT;

    // Workspace layout (floats): h | t0 | t1 | degS/invS | degR/invR
    float* ws   = (float*)d_ws;
    float* h    = ws;
    float* t0   = ws + NL;
    float* t1   = ws + 2 * NL;
    float* degS = ws + 3 * NL;   // becomes inv_sqrt_s
    float* degR = degS + n;      // becomes inv_sqrt_r

    const int feTotal = n * LATENT;

    // Degrees -> inv_sqrt factors (degS & degR are contiguous: zero/finalize together)
    gcn_fill_zero<<<(2 * n + 255) / 256, 256, 0, stream>>>(degS, 2 * n);
    gcn_deg_accum<<<(nE + 255) / 256, 256, 0, stream>>>(senders, recvers, degS, degR, nE);
    gcn_inv_sqrt<<<(2 * n + 255) / 256, 256, 0, stream>>>(degS, 2 * n);

    // Embed
    gcn_embed<<<(feTotal + 255) / 256, 256, 0, stream>>>(nodes, W_embed, b_embed, h, n);

    const int gemmBlocks = (n + 127) / 128;
    for (int step = 0; step < STEPS; ++step) {
        const float* W0 = mlp_W + ((size_t)step * 2 + 0) * LATENT * LATENT;
        const float* W1 = mlp_W + ((size_t)step * 2 + 1) * LATENT * LATENT;
        const float* b0 = mlp_b + ((size_t)step * 2 + 0) * LATENT;
        const float* b1 = mlp_b + ((size_t)step * 2 + 1) * LATENT;

        // x = relu(h @ W0 + b0)
        gcn_mlp_gemm_relu<<<gemmBlocks, 256, 0, stream>>>(h, W0, b0, nullptr, t0, n);
        // x = relu(x @ W1 + b1) * inv_sqrt_s   (scale fused into epilogue)
        gcn_mlp_gemm_relu<<<gemmBlocks, 256, 0, stream>>>(t0, W1, b1, degS, t1, n);

        // agg = scatter_sum(x[senders] -> receivers)
        gcn_fill_zero<<<(feTotal + 255) / 256, 256, 0, stream>>>(t0, feTotal);
        gcn_scatter<<<(nE + 7) / 8, 256, 0, stream>>>(t1, senders, recvers, t0, nE);

        // h = LN(h + agg * inv_sqrt_r)
        gcn_update_ln<<<(n + 7) / 8, 256, 0, stream>>>(
            h, t0, degR, ln_scale + (size_t)step * LATENT, ln_bias + (size_t)step * LATENT, n);
    }

    // Decode
    gcn_decode<<<(n + 7) / 8, 256, 0, stream>>>(h, W_dec, b_dec, out, n);
}